// SimpleAttentionPool_74277164417599
// MI455X (gfx1250) — compile-verified
//
#include <hip/hip_runtime.h>
#include <math.h>

#define NSEG     512
#define FDIM     256
#define FV       (FDIM / 4)   // 64 float4 per row
#define ROWS_PAR 4            // row-groups processed in parallel per block
#define CROWS    56           // rows of the segment cached in LDS (56 KB)
#define EPS_F    1e-16f

// Match the builtin's parameter type exactly: int vector_size(16), AS1/AS3.
typedef int v4i __attribute__((vector_size(16)));
typedef __attribute__((address_space(1))) v4i as1_v4i;
typedef __attribute__((address_space(3))) v4i as3_v4i;
typedef __attribute__((address_space(3))) void as3_void;

// Async DMA: one float4 per lane, global -> LDS, tracked by ASYNCcnt.
__device__ __forceinline__ void async_copy_f4(const float4* gsrc, float4* ldst) {
#if __has_builtin(__builtin_amdgcn_global_load_async_to_lds_b128)
    __builtin_amdgcn_global_load_async_to_lds_b128(
        (as1_v4i*)gsrc, (as3_v4i*)ldst, /*offset=*/0, /*cpol=*/0);
#else
    unsigned lds_off = (unsigned)(size_t)(as3_void*)ldst;
    asm volatile("global_load_async_to_lds_b128 %0, %1, off"
                 :: "v"(lds_off), "v"((const void*)gsrc) : "memory");
#endif
}

__device__ __forceinline__ void wait_async0() {
#if __has_builtin(__builtin_amdgcn_s_wait_asynccnt)
    __builtin_amdgcn_s_wait_asynccnt(0);
#else
    asm volatile("s_wait_asynccnt 0x0" ::: "memory");
#endif
    asm volatile("" ::: "memory");   // keep LDS reads below the wait
}

// Uniform (wave-scalar) binary search over the sorted segment-id array.
__device__ __forceinline__ int lower_bound_i32(const int* __restrict__ a, int n, int key) {
    int lo = 0, hi = n;
    while (lo < hi) {
        int mid = (lo + hi) >> 1;
        if (a[mid] < key) lo = mid + 1; else hi = mid;
    }
    return lo;
}

__global__ __launch_bounds__(FV * ROWS_PAR)
void seg_softmax_kernel(const float* __restrict__ x1, const int* __restrict__ b1,
                        const float* __restrict__ x2, const int* __restrict__ b2,
                        float* __restrict__ out, int N) {
    const int  seg    = blockIdx.x;
    const bool second = (blockIdx.y != 0);

    const float4* __restrict__ xv  = (const float4*)(second ? x2 : x1);
    const int*    __restrict__ bat = second ? b2 : b1;
    float4*       __restrict__ ov  = (float4*)out + (second ? (size_t)N * FV : (size_t)0);

    const int tid = threadIdx.x;
    const int c   = tid & (FV - 1);   // float4 column 0..63
    const int j   = tid >> 6;         // row-group 0..3

    // Segment row range [lo, hi) — uniform across the block.
    const int lo = lower_bound_i32(bat, N, seg);
    const int hi = lower_bound_i32(bat, N, seg + 1);
    if (lo >= hi) return;             // empty segment: no output rows exist

    __shared__ float4 cache[CROWS][FV];   // LDS-resident prefix of the segment
    __shared__ float4 red[ROWS_PAR][FV];  // cross-row-group reduction buffer

    const int len    = hi - lo;
    const int cached = (len < CROWS) ? len : CROWS;

    // Kick off the async DMA fill of the LDS prefix (overlaps with pass 1a).
    for (int rl = j; rl < cached; rl += ROWS_PAR)
        async_copy_f4(&xv[(size_t)(lo + rl) * FV + c], &cache[rl][c]);

    // ---------------- Pass 1a: max over the non-cached suffix (HBM stream) --
    float4 m;
    m.x = m.y = m.z = m.w = -3.402823466e+38f;
    for (int r = lo + cached + j; r < hi; r += ROWS_PAR) {
        __builtin_prefetch((const void*)&xv[(size_t)(r + 4 * ROWS_PAR) * FV + c], 0, 0);
        float4 v = xv[(size_t)r * FV + c];
        m.x = fmaxf(m.x, v.x); m.y = fmaxf(m.y, v.y);
        m.z = fmaxf(m.z, v.z); m.w = fmaxf(m.w, v.w);
    }
    // ---------------- Pass 1b: max over the LDS prefix ---------------------
    wait_async0();   // each thread consumes exactly its own async-loaded data
    for (int rl = j; rl < cached; rl += ROWS_PAR) {
        float4 v = cache[rl][c];
        m.x = fmaxf(m.x, v.x); m.y = fmaxf(m.y, v.y);
        m.z = fmaxf(m.z, v.z); m.w = fmaxf(m.w, v.w);
    }
    red[j][c] = m;
    __syncthreads();
    m = red[0][c];
    #pragma unroll
    for (int k = 1; k < ROWS_PAR; ++k) {
        float4 t = red[k][c];
        m.x = fmaxf(m.x, t.x); m.y = fmaxf(m.y, t.y);
        m.z = fmaxf(m.z, t.z); m.w = fmaxf(m.w, t.w);
    }
    __syncthreads();

    // ---------------- Pass 2: sum of exp(x - m) (LDS prefix + L2 suffix) ---
    float4 s;
    s.x = s.y = s.z = s.w = 0.0f;
    for (int rl = j; rl < cached; rl += ROWS_PAR) {
        float4 v = cache[rl][c];
        s.x += expf(v.x - m.x); s.y += expf(v.y - m.y);
        s.z += expf(v.z - m.z); s.w += expf(v.w - m.w);
    }
    for (int r = lo + cached + j; r < hi; r += ROWS_PAR) {
        float4 v = xv[(size_t)r * FV + c];
        s.x += expf(v.x - m.x); s.y += expf(v.y - m.y);
        s.z += expf(v.z - m.z); s.w += expf(v.w - m.w);
    }
    red[j][c] = s;
    __syncthreads();
    s.x = s.y = s.z = s.w = 0.0f;
    #pragma unroll
    for (int k = 0; k < ROWS_PAR; ++k) {
        float4 t = red[k][c];
        s.x += t.x; s.y += t.y; s.z += t.z; s.w += t.w;
    }

    float4 inv;
    inv.x = 1.0f / (s.x + EPS_F); inv.y = 1.0f / (s.y + EPS_F);
    inv.z = 1.0f / (s.z + EPS_F); inv.w = 1.0f / (s.w + EPS_F);

    // ---------------- Pass 3: normalize + single HBM write per element -----
    for (int rl = j; rl < cached; rl += ROWS_PAR) {
        float4 v = cache[rl][c];
        float4 o;
        o.x = expf(v.x - m.x) * inv.x; o.y = expf(v.y - m.y) * inv.y;
        o.z = expf(v.z - m.z) * inv.z; o.w = expf(v.w - m.w) * inv.w;
        ov[(size_t)(lo + rl) * FV + c] = o;
    }
    for (int r = lo + cached + j; r < hi; r += ROWS_PAR) {
        float4 v = xv[(size_t)r * FV + c];
        float4 o;
        o.x = expf(v.x - m.x) * inv.x; o.y = expf(v.y - m.y) * inv.y;
        o.z = expf(v.z - m.z) * inv.z; o.w = expf(v.w - m.w) * inv.w;
        ov[(size_t)r * FV + c] = o;
    }
}

extern "C" void kernel_launch(void* const* d_in, const int* in_sizes, int n_in,
                              void* d_out, int out_size, void* d_ws, size_t ws_size,
                              hipStream_t stream) {
    const float* x1 = (const float*)d_in[0];
    const int*   b1 = (const int*)d_in[1];
    const float* x2 = (const float*)d_in[2];
    const int*   b2 = (const int*)d_in[3];
    float*       out = (float*)d_out;

    const int N = in_sizes[1];   // rows (batch_1 has one id per row)

    dim3 grid(NSEG, 2);
    dim3 block(FV * ROWS_PAR);   // 256 threads = 8 wave32
    seg_softmax_kernel<<<grid, block, 0, stream>>>(x1, b1, x2, b2, out, N);
}